// CNN_8134668058969
// MI455X (gfx1250) — compile-verified
//
#include <hip/hip_runtime.h>
#include <hip/hip_bf16.h>

typedef float v2f __attribute__((ext_vector_type(2)));
typedef float v8f __attribute__((ext_vector_type(8)));

#define NEG_SLOPE 0.2f

// ---------------------------------------------------------------- utilities
__device__ __forceinline__ void atomAddF32(float* p, float v) {
    // lowers to global_atomic_add_f32 (no return)
    __hip_atomic_fetch_add(p, v, __ATOMIC_RELAXED, __HIP_MEMORY_SCOPE_AGENT);
}

__device__ __forceinline__ void atomMaxF32(float* p, float v) {
    // sign-aware int punning: works for mixed-sign floats, init must be -inf
    if (v >= 0.0f) atomicMax((int*)p, __float_as_int(v));
    else           atomicMin((unsigned int*)p, __float_as_uint(v));
}

__device__ __forceinline__ float lrelu(float v) {
    return v > 0.0f ? v : NEG_SLOPE * v;
}

// ---------------------------------------------------------------- init
// agg1[n,f] = b1[f]; out[n,c] = b2[c]; m=-inf; denom=0
__global__ __launch_bounds__(256) void init_kernel(
    float* __restrict__ agg1, float* __restrict__ out,
    const float* __restrict__ b1, const float* __restrict__ b2,
    float* __restrict__ m1, float* __restrict__ dn1,
    float* __restrict__ m2, float* __restrict__ dn2, int N)
{
    long i = (long)blockIdx.x * 256 + threadIdx.x;
    long tot = (long)N * 128;
    if (i < tot)        agg1[i] = b1[i & 127];
    if (i < (long)N*16) out[i]  = b2[i & 15];
    if (i < N) {
        float ninf = -__builtin_inff();
        m1[i] = ninf; m2[i] = ninf;
        dn1[i] = 0.0f; dn2[i] = 0.0f;
    }
}

// ---------------------------------------------------------------- GEMM1: h1 = x @ W1  (N x 128 @ 128 x 128), fp32 WMMA
__global__ __launch_bounds__(256) void gemm1_wmma(
    const float* __restrict__ x, const float* __restrict__ W1,
    float* __restrict__ h1, int N)
{
    __shared__ float w1s[128 * 128];  // 64 KB of the 320 KB WGP LDS
    for (int i = threadIdx.x; i < (128 * 128) / 4; i += 256)
        ((float4*)w1s)[i] = ((const float4*)W1)[i];
    __syncthreads();

    const int wave = threadIdx.x >> 5;
    const int lane = threadIdx.x & 31;
    const int hf   = lane >> 4;        // lane half selects K pair (A) / K row (B)
    const int lrow = lane & 15;

    const int rowBase = blockIdx.x * 128 + wave * 16;
    int arow = rowBase + lrow;
    if (arow > N - 1) arow = N - 1;    // clamp: keep EXEC all-ones for WMMA
    const float* xrow = x + (size_t)arow * 128 + 2 * hf;

    v8f acc[8];
#pragma unroll
    for (int t = 0; t < 8; ++t) acc[t] = (v8f){};

    for (int k0 = 0; k0 < 128; k0 += 4) {
        v2f a;                                    // A 16x4 f32 frag: 2 VGPRs
        a.x = xrow[k0];
        a.y = xrow[k0 + 1];
        const int kb = k0 + 2 * hf;
#pragma unroll
        for (int ct = 0; ct < 8; ++ct) {          // 8 column tiles of 16
            const int col = ct * 16 + lrow;
            v2f b;                                // B 4x16 f32 frag from LDS
            b.x = w1s[kb * 128 + col];
            b.y = w1s[(kb + 1) * 128 + col];
            acc[ct] = __builtin_amdgcn_wmma_f32_16x16x4_f32(
                false, a, false, b, (short)0, acc[ct], false, false);
        }
    }

#pragma unroll
    for (int r = 0; r < 8; ++r) {
        const int row = rowBase + r + hf * 8;     // C layout: VGPR r -> M=r (+8 for hi half)
        if (row < N) {
#pragma unroll
            for (int ct = 0; ct < 8; ++ct)
                h1[(size_t)row * 128 + ct * 16 + lrow] = acc[ct][r];
        }
    }
}

// ---------------------------------------------------------------- GEMM2: h2 = relu(agg1) @ W2  (N x 128 @ 128 x 16)
__global__ __launch_bounds__(256) void gemm2_wmma(
    const float* __restrict__ agg1, const float* __restrict__ W2,
    float* __restrict__ h2, int N)
{
    __shared__ float w2s[128 * 16];   // 8 KB
    for (int i = threadIdx.x; i < (128 * 16) / 4; i += 256)
        ((float4*)w2s)[i] = ((const float4*)W2)[i];
    __syncthreads();

    const int wave = threadIdx.x >> 5;
    const int lane = threadIdx.x & 31;
    const int hf   = lane >> 4;
    const int lrow = lane & 15;

    const int rowBase = blockIdx.x * 128 + wave * 16;
    int arow = rowBase + lrow;
    if (arow > N - 1) arow = N - 1;
    const float* ap = agg1 + (size_t)arow * 128 + 2 * hf;

    v8f acc = (v8f){};
    for (int k0 = 0; k0 < 128; k0 += 4) {
        v2f a;
        a.x = fmaxf(ap[k0], 0.0f);        // fused ReLU on the A operand
        a.y = fmaxf(ap[k0 + 1], 0.0f);
        const int kb = k0 + 2 * hf;
        v2f b;
        b.x = w2s[kb * 16 + lrow];
        b.y = w2s[(kb + 1) * 16 + lrow];
        acc = __builtin_amdgcn_wmma_f32_16x16x4_f32(
            false, a, false, b, (short)0, acc, false, false);
    }

#pragma unroll
    for (int r = 0; r < 8; ++r) {
        const int row = rowBase + r + hf * 8;
        if (row < N) h2[(size_t)row * 16 + lrow] = acc[r];
    }
}

// ---------------------------------------------------------------- alpha (layer 1): wave32 per node, 128-wide dot
__global__ __launch_bounds__(256) void alpha1_kernel(
    const float* __restrict__ h1, const float* __restrict__ a_src,
    const float* __restrict__ a_dst, float* __restrict__ as,
    float* __restrict__ ad, int N)
{
    const int n    = (int)(((long)blockIdx.x * 256 + threadIdx.x) >> 5);
    const int lane = threadIdx.x & 31;
    if (n >= N) return;
    float4 v  = ((const float4*)(h1 + (size_t)n * 128))[lane];
    float4 s4 = ((const float4*)a_src)[lane];
    float4 d4 = ((const float4*)a_dst)[lane];
    float ps = v.x * s4.x + v.y * s4.y + v.z * s4.z + v.w * s4.w;
    float pd = v.x * d4.x + v.y * d4.y + v.z * d4.z + v.w * d4.w;
#pragma unroll
    for (int off = 16; off > 0; off >>= 1) {
        ps += __shfl_xor(ps, off, 32);
        pd += __shfl_xor(pd, off, 32);
    }
    if (lane == 0) { as[n] = ps; ad[n] = pd; }
}

// ---------------------------------------------------------------- alpha (layer 2): thread per node, 16-wide dot
__global__ __launch_bounds__(256) void alpha2_kernel(
    const float* __restrict__ h2, const float* __restrict__ a_src,
    const float* __restrict__ a_dst, float* __restrict__ as,
    float* __restrict__ ad, int N)
{
    const int n = (int)((long)blockIdx.x * 256 + threadIdx.x);
    if (n >= N) return;
    const float* r = h2 + (size_t)n * 16;
    float ps = 0.0f, pd = 0.0f;
#pragma unroll
    for (int c = 0; c < 16; ++c) {
        const float v = r[c];
        ps += v * a_src[c];
        pd += v * a_dst[c];
    }
    as[n] = ps; ad[n] = pd;
}

// ---------------------------------------------------------------- segment max over dst
__global__ __launch_bounds__(256) void edge_max_kernel(
    const long long* __restrict__ src, const long long* __restrict__ dst,
    long E, long Etot, const float* __restrict__ as,
    const float* __restrict__ ad, float* __restrict__ m)
{
    const long e = (long)blockIdx.x * 256 + threadIdx.x;
    if (e >= Etot) return;
    long s, d;
    if (e < E) { s = (long)src[e]; d = (long)dst[e]; }
    else       { s = d = e - E; }                       // self loop
    atomMaxF32(&m[d], lrelu(as[s] + ad[d]));
}

// ---------------------------------------------------------------- segment sum of exp(e - max)
__global__ __launch_bounds__(256) void edge_sum_kernel(
    const long long* __restrict__ src, const long long* __restrict__ dst,
    long E, long Etot, const float* __restrict__ as,
    const float* __restrict__ ad, const float* __restrict__ m,
    float* __restrict__ dn)
{
    const long e = (long)blockIdx.x * 256 + threadIdx.x;
    if (e >= Etot) return;
    long s, d;
    if (e < E) { s = (long)src[e]; d = (long)dst[e]; }
    else       { s = d = e - E; }
    const float lr = lrelu(as[s] + ad[d]);
    atomAddF32(&dn[d], __expf(lr - m[d]));
}

// ---------------------------------------------------------------- layer-1 scatter: wave32 per edge, 128 features (float4/lane)
__global__ __launch_bounds__(256) void scatter1_kernel(
    const long long* __restrict__ src, const long long* __restrict__ dst,
    long E, long Etot, const float* __restrict__ as,
    const float* __restrict__ ad, const float* __restrict__ m,
    const float* __restrict__ dn, const float* __restrict__ h1,
    float* __restrict__ agg)
{
    const long we  = ((long)blockIdx.x * 256 + threadIdx.x) >> 5;
    const int lane = threadIdx.x & 31;
    if (we >= Etot) return;
    long s, d;
    if (we < E) { s = (long)src[we]; d = (long)dst[we]; }
    else        { s = d = we - E; }
    const float lr = lrelu(as[s] + ad[d]);
    const float w  = __expf(lr - m[d]) / dn[d];
    const float4 v = ((const float4*)(h1 + (size_t)s * 128))[lane];
    float* op = agg + (size_t)d * 128 + lane * 4;
    atomAddF32(op + 0, v.x * w);
    atomAddF32(op + 1, v.y * w);
    atomAddF32(op + 2, v.z * w);
    atomAddF32(op + 3, v.w * w);
}

// ---------------------------------------------------------------- layer-2 scatter: 16 threads per edge
__global__ __launch_bounds__(256) void scatter2_kernel(
    const long long* __restrict__ src, const long long* __restrict__ dst,
    long E, long Etot, const float* __restrict__ as,
    const float* __restrict__ ad, const float* __restrict__ m,
    const float* __restrict__ dn, const float* __restrict__ h2,
    float* __restrict__ out)
{
    const long t = (long)blockIdx.x * 256 + threadIdx.x;
    const long e = t >> 4;
    const int  c = (int)(t & 15);
    if (e >= Etot) return;
    long s, d;
    if (e < E) { s = (long)src[e]; d = (long)dst[e]; }
    else       { s = d = e - E; }
    const float lr = lrelu(as[s] + ad[d]);
    const float w  = __expf(lr - m[d]) / dn[d];
    atomAddF32(&out[(size_t)d * 16 + c], h2[(size_t)s * 16 + c] * w);
}

// ---------------------------------------------------------------- launcher
extern "C" void kernel_launch(void* const* d_in, const int* in_sizes, int n_in,
                              void* d_out, int out_size, void* d_ws, size_t ws_size,
                              hipStream_t stream)
{
    const float*     x   = (const float*)d_in[0];
    const long long* ei  = (const long long*)d_in[1];   // int64 [2, E]
    const float*     W1  = (const float*)d_in[2];
    const float*     as1w = (const float*)d_in[3];
    const float*     ad1w = (const float*)d_in[4];
    const float*     b1  = (const float*)d_in[5];
    const float*     W2  = (const float*)d_in[6];
    const float*     as2w = (const float*)d_in[7];
    const float*     ad2w = (const float*)d_in[8];
    const float*     b2  = (const float*)d_in[9];
    float*           out = (float*)d_out;

    const int  N    = in_sizes[0] / 128;
    const long E    = in_sizes[1] / 2;
    const long Etot = E + N;
    const long long* srcI = ei;
    const long long* dstI = ei + E;

    // workspace layout (floats)
    float* ws   = (float*)d_ws;
    float* h1   = ws;                      // N*128
    float* agg1 = h1 + (size_t)N * 128;    // N*128
    float* h2   = agg1 + (size_t)N * 128;  // N*16
    float* as1  = h2 + (size_t)N * 16;     // N
    float* ad1  = as1 + N;
    float* m1   = ad1 + N;
    float* dn1  = m1 + N;
    float* as2  = dn1 + N;
    float* ad2  = as2 + N;
    float* m2   = ad2 + N;
    float* dn2  = m2 + N;

    const int bInit  = (int)(((long)N * 128 + 255) / 256);
    const int bGemm  = (N + 127) / 128;
    const int bWaveN = (N + 7) / 8;          // wave32 per node
    const int bNode  = (N + 255) / 256;
    const int bEdge  = (int)((Etot + 255) / 256);
    const int bWaveE = (int)((Etot + 7) / 8);  // wave32 per edge
    const int bE16   = (int)((Etot * 16 + 255) / 256);

    init_kernel<<<bInit, 256, 0, stream>>>(agg1, out, b1, b2, m1, dn1, m2, dn2, N);

    // ---- layer 1
    gemm1_wmma<<<bGemm, 256, 0, stream>>>(x, W1, h1, N);
    alpha1_kernel<<<bWaveN, 256, 0, stream>>>(h1, as1w, ad1w, as1, ad1, N);
    edge_max_kernel<<<bEdge, 256, 0, stream>>>(srcI, dstI, E, Etot, as1, ad1, m1);
    edge_sum_kernel<<<bEdge, 256, 0, stream>>>(srcI, dstI, E, Etot, as1, ad1, m1, dn1);
    scatter1_kernel<<<bWaveE, 256, 0, stream>>>(srcI, dstI, E, Etot, as1, ad1, m1, dn1, h1, agg1);

    // ---- layer 2 (ReLU fused into GEMM2 A-operand load)
    gemm2_wmma<<<bGemm, 256, 0, stream>>>(agg1, W2, h2, N);
    alpha2_kernel<<<bNode, 256, 0, stream>>>(h2, as2w, ad2w, as2, ad2, N);
    edge_max_kernel<<<bEdge, 256, 0, stream>>>(srcI, dstI, E, Etot, as2, ad2, m2);
    edge_sum_kernel<<<bEdge, 256, 0, stream>>>(srcI, dstI, E, Etot, as2, ad2, m2, dn2);
    scatter2_kernel<<<bE16, 256, 0, stream>>>(srcI, dstI, E, Etot, as2, ad2, m2, dn2, h2, out);
}